// ZeroOneLoss_41094247088883
// MI455X (gfx1250) — compile-verified
//
#include <hip/hip_runtime.h>

// ZeroOneLoss: out = mean_i( input[i, targets[i]] > 0 )
// B = 16384, V = 32000 (derived from in_sizes at launch time).
//
// Latency-bound gather + reduction. Strategy:
//   - 1 thread per row -> 1 coalesced target load + 1 scattered gather.
//   - wave32 ballot + popcount for the per-wave reduction (cheapest lane
//     reduction on gfx1250), LDS combine across the 8 waves of a block.
//   - deterministic two-pass finish: per-block u32 partials in d_ws,
//     single-block finalize writes count / B. No atomics, no float
//     accumulation-order sensitivity, no dependence on buffer pre-zeroing.

#define ZOL_THREADS 256
#define ZOL_WAVES   (ZOL_THREADS / 32)   // wave32 on gfx1250

__global__ void __launch_bounds__(ZOL_THREADS)
zol_partial_kernel(const float* __restrict__ input,
                   const int* __restrict__ targets,
                   unsigned int* __restrict__ partials,
                   int B, int V) {
    const int tid = blockIdx.x * ZOL_THREADS + threadIdx.x;

    bool pred = false;
    if (tid < B) {
        const int t = targets[tid];                       // coalesced load
        const size_t idx = (size_t)tid * (size_t)V + (size_t)t;
        pred = input[idx] > 0.0f;                         // scattered gather
    }

    // wave32 ballot: v_cmp -> s_bcnt over the 32-bit mask
    const unsigned long long mask = __ballot(pred);
    const unsigned int wave_cnt = (unsigned int)__popcll(mask);

    __shared__ unsigned int s_wave[ZOL_WAVES];
    const int lane = threadIdx.x & 31;
    const int wave = threadIdx.x >> 5;
    if (lane == 0) s_wave[wave] = wave_cnt;
    __syncthreads();

    if (threadIdx.x == 0) {
        unsigned int s = 0;
        #pragma unroll
        for (int i = 0; i < ZOL_WAVES; ++i) s += s_wave[i];
        partials[blockIdx.x] = s;                         // always rewritten
    }
}

__global__ void __launch_bounds__(ZOL_THREADS)
zol_finalize_kernel(const unsigned int* __restrict__ partials,
                    float* __restrict__ out,
                    int nblocks, int B) {
    // Single-block deterministic reduction of the per-block partials.
    unsigned int acc = 0;
    for (int i = threadIdx.x; i < nblocks; i += ZOL_THREADS)
        acc += partials[i];

    __shared__ unsigned int s_wave[ZOL_WAVES];
    // reduce within wave via ballot-free shfl tree (wave32)
    for (int off = 16; off > 0; off >>= 1)
        acc += __shfl_down(acc, off, 32);

    const int lane = threadIdx.x & 31;
    const int wave = threadIdx.x >> 5;
    if (lane == 0) s_wave[wave] = acc;
    __syncthreads();

    if (threadIdx.x == 0) {
        unsigned int s = 0;
        #pragma unroll
        for (int i = 0; i < ZOL_WAVES; ++i) s += s_wave[i];
        // counts <= B = 16384 are exact in fp32
        out[0] = (float)s / (float)B;
    }
}

extern "C" void kernel_launch(void* const* d_in, const int* in_sizes, int n_in,
                              void* d_out, int out_size, void* d_ws, size_t ws_size,
                              hipStream_t stream) {
    const float* input   = (const float*)d_in[0];   // [B, V] f32
    const int*   targets = (const int*)d_in[1];     // [B] int
    float*       out     = (float*)d_out;           // scalar f32

    const int B = in_sizes[1];
    const int V = in_sizes[0] / B;

    unsigned int* partials = (unsigned int*)d_ws;
    const int nblocks = (B + ZOL_THREADS - 1) / ZOL_THREADS;   // 64 for B=16384

    zol_partial_kernel<<<nblocks, ZOL_THREADS, 0, stream>>>(input, targets,
                                                            partials, B, V);
    zol_finalize_kernel<<<1, ZOL_THREADS, 0, stream>>>(partials, out,
                                                       nblocks, B);
}